// GATEncoder_62517543961154
// MI455X (gfx1250) — compile-verified
//
#include <hip/hip_runtime.h>
#include <hip/hip_bf16.h>

#define HID   128
#define HEADS 4
#define PH    32
#define BN_EPS 1e-5f
#define SLOPE  0.2f

typedef float v2f __attribute__((ext_vector_type(2)));
typedef float v8f __attribute__((ext_vector_type(8)));

// ---- order-preserving float<->int encoding (for atomicMax on signed int) ----
__device__ __forceinline__ int f2ord(float f) {
    int b = __float_as_int(f);
    return b >= 0 ? b : (b ^ 0x7FFFFFFF);
}
__device__ __forceinline__ float ord2f(int k) {
    return __int_as_float(k >= 0 ? k : (k ^ 0x7FFFFFFF));
}

// ---------------------------------------------------------------------------
// C[nrows,128] = A[nrows,128] @ B[128,128] (+ optional bias), FP32 WMMA.
// Block = 256 threads = 8 waves; block computes a 16-row x 128-col tile,
// wave w owns the 16x16 tile at columns [16w, 16w+16).
// A-frag (16x4 f32, ISA 7.12.2): lanes 0-15 -> M=lane, VGPR0=K0 VGPR1=K1;
//                                lanes 16-31 -> M=lane-16, VGPR0=K2 VGPR1=K3.
// B-frag (4x16): same striping with N=lane&15.
// C/D: VGPR v -> row M=v (lanes 0-15) / M=v+8 (lanes 16-31), N=lane&15.
// ---------------------------------------------------------------------------
__global__ void gemm128_wmma(const float* __restrict__ A,
                             const float* __restrict__ B,
                             const float* __restrict__ bias,
                             float* __restrict__ C, int nrows) {
    const int wave  = threadIdx.x >> 5;
    const int lane  = threadIdx.x & 31;
    const int l16   = lane & 15;
    const int lhalf = lane >> 4;
    const int rowBase = blockIdx.x << 4;
    const int colBase = wave << 4;

    const float* Arow = A + (size_t)(rowBase + l16) * HID;
    const float* Bcol = B + colBase + l16;

    v8f acc = {0.f, 0.f, 0.f, 0.f, 0.f, 0.f, 0.f, 0.f};
#pragma unroll
    for (int k = 0; k < HID; k += 4) {
        const int ka = k + 2 * lhalf;
        v2f a, b;
        a.x = Arow[ka];
        a.y = Arow[ka + 1];
        b.x = Bcol[(size_t)ka * HID];
        b.y = Bcol[(size_t)(ka + 1) * HID];
        acc = __builtin_amdgcn_wmma_f32_16x16x4_f32(
            /*neg_a=*/false, a, /*neg_b=*/false, b,
            /*c_mod=*/(short)0, acc, /*reuse_a=*/false, /*reuse_b=*/false);
    }
    const float bv = bias ? bias[colBase + l16] : 0.f;
#pragma unroll
    for (int v = 0; v < 8; ++v) {
        const int row = rowBase + v + 8 * lhalf;
        C[(size_t)row * HID + colBase + l16] = acc[v] + bv;
    }
}

// ---- per-column sum / sumsq over rows (stats must be pre-zeroed) ----------
__global__ void colstats(const float* __restrict__ X, float* __restrict__ stats,
                         int nrows, int rowsPerBlock) {
    const int c    = threadIdx.x & 127;
    const int rsub = threadIdx.x >> 7;          // 0 or 1
    int r  = blockIdx.x * rowsPerBlock + rsub;
    int rEnd = blockIdx.x * rowsPerBlock + rowsPerBlock;
    if (rEnd > nrows) rEnd = nrows;
    float s = 0.f, q = 0.f;
    for (; r < rEnd; r += 2) {
        float x = X[(size_t)r * HID + c];
        s += x; q += x * x;
    }
    __shared__ float sh[256], qh[256];
    sh[threadIdx.x] = s; qh[threadIdx.x] = q;
    __syncthreads();
    if (rsub == 0) {
        s += sh[threadIdx.x + 128];
        q += qh[threadIdx.x + 128];
        atomicAdd(&stats[c], s);
        atomicAdd(&stats[HID + c], q);
    }
}

// ---- BN (training-mode, biased var) + ELU, optional residual add ----------
__global__ void bn_elu(const float* __restrict__ X, const float* __restrict__ stats,
                       const float* __restrict__ gamma, const float* __restrict__ beta,
                       float* __restrict__ H, int total, float invN, int residual) {
    const int idx = blockIdx.x * 256 + threadIdx.x;
    if (idx >= total) return;
    const int c = idx & 127;
    const float mean = stats[c] * invN;
    const float var  = stats[HID + c] * invN - mean * mean;
    float v = gamma[c] * (X[idx] - mean) * rsqrtf(var + BN_EPS) + beta[c];
    v = v > 0.f ? v : expm1f(v);                 // ELU (alpha = 1)
    H[idx] = residual ? H[idx] + v : v;
}

// ---- per-node attention scores: a_src/a_dst [N, HEADS] --------------------
__global__ void attn_scores(const float* __restrict__ xh,
                            const float* __restrict__ att_s,
                            const float* __restrict__ att_d,
                            float* __restrict__ a_src, float* __restrict__ a_dst,
                            int nrows) {
    const int n = blockIdx.x * 256 + threadIdx.x;
    if (n >= nrows) return;
    const float4* row = (const float4*)(xh + (size_t)n * HID);
    const float4* as4 = (const float4*)att_s;
    const float4* ad4 = (const float4*)att_d;
    float s[HEADS] = {0.f, 0.f, 0.f, 0.f};
    float d[HEADS] = {0.f, 0.f, 0.f, 0.f};
#pragma unroll
    for (int q = 0; q < 32; ++q) {               // 32 float4 = 128 cols
        const float4 x = row[q];
        const float4 a = as4[q];
        const float4 b = ad4[q];
        const int h = q >> 3;                    // 8 float4 per head
        s[h] += x.x * a.x + x.y * a.y + x.z * a.z + x.w * a.w;
        d[h] += x.x * b.x + x.y * b.y + x.z * b.z + x.w * b.w;
    }
#pragma unroll
    for (int h = 0; h < HEADS; ++h) {
        a_src[n * HEADS + h] = s[h];
        a_dst[n * HEADS + h] = d[h];
    }
}

// ---- init per-layer scratch ----------------------------------------------
__global__ void init_layer(float* __restrict__ acc, int* __restrict__ mkey,
                           float* __restrict__ denom, float* __restrict__ stats,
                           int nrows) {
    const int idx = blockIdx.x * 256 + threadIdx.x;
    if (idx < nrows * HID)  acc[idx] = 0.f;
    if (idx < nrows * HEADS) { mkey[idx] = (int)0x80000000; denom[idx] = 0.f; }
    if (idx < 2 * HID) stats[idx] = 0.f;
}

__global__ void zero_stats(float* __restrict__ stats) { stats[threadIdx.x] = 0.f; }

__device__ __forceinline__ void edge_endpoints(const int* __restrict__ ei, int e,
                                               int E, int& s, int& d) {
    if (e < E) { s = ei[e]; d = ei[E + e]; }
    else       { s = d = e - E; }                // implicit self-loop
}

// ---- pass 1: segment max of leaky_relu(a_src[s]+a_dst[d]) over dst --------
__global__ void edge_max(const int* __restrict__ ei,
                         const float* __restrict__ a_src,
                         const float* __restrict__ a_dst,
                         int* __restrict__ mkey, int E, int Etot) {
    const int e = blockIdx.x * 256 + threadIdx.x;
    if (e >= Etot) return;
    int s, d; edge_endpoints(ei, e, E, s, d);
#pragma unroll
    for (int h = 0; h < HEADS; ++h) {
        float v = a_src[s * HEADS + h] + a_dst[d * HEADS + h];
        v = v > 0.f ? v : SLOPE * v;
        atomicMax(&mkey[d * HEADS + h], f2ord(v));
    }
}

// ---- pass 2: ex = exp(e - max); denom = segment_sum(ex) -------------------
__global__ void edge_expsum(const int* __restrict__ ei,
                            const float* __restrict__ a_src,
                            const float* __restrict__ a_dst,
                            const int* __restrict__ mkey,
                            float* __restrict__ denom, float* __restrict__ ex,
                            int E, int Etot) {
    const int e = blockIdx.x * 256 + threadIdx.x;
    if (e >= Etot) return;
    int s, d; edge_endpoints(ei, e, E, s, d);
#pragma unroll
    for (int h = 0; h < HEADS; ++h) {
        float v = a_src[s * HEADS + h] + a_dst[d * HEADS + h];
        v = v > 0.f ? v : SLOPE * v;
        const float m = ord2f(mkey[d * HEADS + h]);
        const float x = expf(v - m);
        ex[e * HEADS + h] = x;
        atomicAdd(&denom[d * HEADS + h], x);
    }
}

// ---- pass 3: acc[dst, c] += xh[src, c] * alpha(e, head(c)) ----------------
__global__ void edge_agg(const int* __restrict__ ei,
                         const float* __restrict__ xh,
                         const float* __restrict__ ex,
                         const float* __restrict__ denom,
                         float* __restrict__ acc, int E, int Etot) {
    const int idx = blockIdx.x * 256 + threadIdx.x;   // Etot*128 < 2^31
    if (idx >= Etot * HID) return;
    const int e = idx >> 7;
    const int c = idx & 127;
    const int h = c >> 5;
    int s, d; edge_endpoints(ei, e, E, s, d);
    const float alpha = ex[e * HEADS + h] / denom[d * HEADS + h];
    atomicAdd(&acc[(size_t)d * HID + c], xh[(size_t)s * HID + c] * alpha);
}

// ---------------------------------------------------------------------------
extern "C" void kernel_launch(void* const* d_in, const int* in_sizes, int n_in,
                              void* d_out, int out_size, void* d_ws, size_t ws_size,
                              hipStream_t stream) {
    (void)n_in; (void)out_size; (void)ws_size;
    const float* x        = (const float*)d_in[0];
    const int*   ei       = (const int*)  d_in[1];
    const float* in_w     = (const float*)d_in[2];
    const float* in_b     = (const float*)d_in[3];
    const float* in_gamma = (const float*)d_in[4];
    const float* in_beta  = (const float*)d_in[5];
    const float* W        = (const float*)d_in[6];
    const float* att_src  = (const float*)d_in[7];
    const float* att_dst  = (const float*)d_in[8];
    // d_in[9] = per-layer bias: per-column constant is cancelled by BN mean-sub.
    const float* gamma    = (const float*)d_in[10];
    const float* beta     = (const float*)d_in[11];
    float* h = (float*)d_out;

    const int N    = in_sizes[0] / HID;
    const int E    = in_sizes[1] / 2;
    const int Etot = E + N;

    // workspace carve-out (256B aligned)
    char* wsb = (char*)d_ws;
    size_t off = 0;
    auto carve = [&](size_t bytes) -> void* {
        void* p = wsb + off;
        off = (off + bytes + 255) & ~(size_t)255;
        return p;
    };
    float* xh    = (float*)carve((size_t)N * HID * 4);
    float* acc   = (float*)carve((size_t)N * HID * 4);
    float* a_s   = (float*)carve((size_t)N * HEADS * 4);
    float* a_d   = (float*)carve((size_t)N * HEADS * 4);
    int*   mkey  = (int*)  carve((size_t)N * HEADS * 4);
    float* denom = (float*)carve((size_t)N * HEADS * 4);
    float* ex    = (float*)carve((size_t)Etot * HEADS * 4);
    float* stats = (float*)carve(2 * HID * 4);

    const float invN = 1.f / (float)N;
    const int totNC      = N * HID;
    const int gemmBlocks = (N + 15) / 16;            // N % 16 == 0 (50000)
    const int ncBlocks   = (totNC + 255) / 256;
    const int nodeBlocks = (N + 255) / 256;
    const int edgeBlocks = (Etot + 255) / 256;
    const int aggBlocks  = (Etot * HID + 255) / 256;
    const int statBlocks = (N + 63) / 64;

    // ---- input stage: h = elu(BN(x @ in_w + in_b)) ----
    zero_stats<<<1, 256, 0, stream>>>(stats);
    gemm128_wmma<<<gemmBlocks, 256, 0, stream>>>(x, in_w, in_b, xh, N);
    colstats<<<statBlocks, 256, 0, stream>>>(xh, stats, N, 64);
    bn_elu<<<ncBlocks, 256, 0, stream>>>(xh, stats, in_gamma, in_beta, h,
                                         totNC, invN, /*residual=*/0);

    // ---- GAT layers ----
    for (int l = 0; l < 2; ++l) {
        gemm128_wmma<<<gemmBlocks, 256, 0, stream>>>(
            h, W + (size_t)l * HID * HID, nullptr, xh, N);
        attn_scores<<<nodeBlocks, 256, 0, stream>>>(
            xh, att_src + l * HEADS * PH, att_dst + l * HEADS * PH, a_s, a_d, N);
        init_layer<<<ncBlocks, 256, 0, stream>>>(acc, mkey, denom, stats, N);
        edge_max<<<edgeBlocks, 256, 0, stream>>>(ei, a_s, a_d, mkey, E, Etot);
        edge_expsum<<<edgeBlocks, 256, 0, stream>>>(ei, a_s, a_d, mkey, denom, ex, E, Etot);
        edge_agg<<<aggBlocks, 256, 0, stream>>>(ei, xh, ex, denom, acc, E, Etot);
        colstats<<<statBlocks, 256, 0, stream>>>(acc, stats, N, 64);
        bn_elu<<<ncBlocks, 256, 0, stream>>>(acc, stats, gamma + l * HID,
                                             beta + l * HID, h, totNC, invN,
                                             /*residual=*/1);
    }
}